// DGDCN_81922206204633
// MI455X (gfx1250) — compile-verified
//
#include <hip/hip_runtime.h>
#include <hip/hip_bf16.h>

typedef float v4f __attribute__((ext_vector_type(4)));
typedef int   v4i __attribute__((ext_vector_type(4)));

// Pointer types the async builtins expect (from hipcc diagnostics: param is
// "int __vector(4) __device__ *"  ==  v4i in address space 1).
typedef __attribute__((address_space(1))) v4i* gv4i_ptr;
typedef __attribute__((address_space(3))) v4i* lv4i_ptr;

#if __has_builtin(__builtin_amdgcn_global_load_async_to_lds_b128) && \
    __has_builtin(__builtin_amdgcn_global_store_async_from_lds_b128) && \
    __has_builtin(__builtin_amdgcn_s_wait_asynccnt)
#define HAVE_ASYNC_LDS 1
#else
#define HAVE_ASYNC_LDS 0
#warning "CDNA5_PROBE: async global<->LDS builtins ABSENT"
#endif

#define BLOCK 256

// One block per output (batch) row.
// out[row, 0:cnt*emb_dim]   = embeddings[start:start+cnt]  (contiguous: positions are sorted)
// out[row, cnt*emb_dim:...] = 0
__global__ __launch_bounds__(BLOCK) void remap_rows_kernel(
    const float* __restrict__ emb,
    const int*   __restrict__ pos,
    int n_valid, int emb_dim, int rowlen,
    float* __restrict__ out)
{
    const int row = (int)blockIdx.x;

    // Two uniform binary searches (lower_bound(row), upper_bound(row)).
    // Addresses/control depend only on blockIdx -> scalar (s_load) path.
    int lo = 0, hi = n_valid;
    while (lo < hi) { int mid = (lo + hi) >> 1; if (pos[mid] <  row) lo = mid + 1; else hi = mid; }
    const int start = lo;
    hi = n_valid;
    while (lo < hi) { int mid = (lo + hi) >> 1; if (pos[mid] <= row) lo = mid + 1; else hi = mid; }
    int cnt = lo - start;
    const int maxslots = rowlen / emb_dim;     // hist_len; reference uses mode="drop"
    if (cnt > maxslots) cnt = maxslots;

    const v4f* __restrict__ src = (const v4f*)(emb + (size_t)start * (size_t)emb_dim);
    v4f* __restrict__ dst = (v4f*)(out + (size_t)row * (size_t)rowlen);
    const int n4 = rowlen >> 2;                // float4s per out row (800)
    const int v4 = (cnt * emb_dim) >> 2;       // valid float4 prefix (emb_dim % 4 == 0)

#if HAVE_ASYNC_LDS
    // CDNA5 path: stage the valid prefix through LDS via the async DMA pipe
    // (ASYNCcnt-tracked, bypasses VGPRs), zero-fill the tail in LDS, then
    // async-store the whole row back out.
    extern __shared__ char smem_raw[];
    float* lbuf = (float*)smem_raw;
    for (int j = threadIdx.x; j < n4; j += BLOCK) {
        if (j < v4) {
            __builtin_amdgcn_global_load_async_to_lds_b128(
                (gv4i_ptr)&src[j],
                (lv4i_ptr)&lbuf[j * 4],
                0, 0);
        } else {
            v4f z = {0.f, 0.f, 0.f, 0.f};
            *(v4f*)&lbuf[j * 4] = z;
        }
    }
    __builtin_amdgcn_s_wait_asynccnt(0);   // our async LDS writes have landed
    __syncthreads();                        // (also covers the ds-store zeros)
    for (int j = threadIdx.x; j < n4; j += BLOCK) {
        __builtin_amdgcn_global_store_async_from_lds_b128(
            (gv4i_ptr)&dst[j],
            (lv4i_ptr)&lbuf[j * 4],
            0, 0);
    }
    __builtin_amdgcn_s_wait_asynccnt(0);
#else
    // Fallback: straight coalesced b128 copy with non-temporal cache policy
    // (output is write-once and bigger than the 192MB L2; input is read-once).
    for (int j = threadIdx.x; j < n4; j += BLOCK) {
        v4f v;
        if (j < v4) {
            v = __builtin_nontemporal_load(&src[j]);
        } else {
            v4f z = {0.f, 0.f, 0.f, 0.f};
            v = z;
        }
        __builtin_nontemporal_store(v, &dst[j]);
    }
#endif
}

extern "C" void kernel_launch(void* const* d_in, const int* in_sizes, int n_in,
                              void* d_out, int out_size, void* d_ws, size_t ws_size,
                              hipStream_t stream) {
    const float* emb = (const float*)d_in[0];
    const int*   pos = (const int*)d_in[1];

    const int n_valid = in_sizes[1];
    const int emb_dim = (n_valid > 0) ? (in_sizes[0] / n_valid) : 1;   // 64
    const int hist    = 50;                       // HIST from reference (device scalar unusable for grid sizing)
    const int rowlen  = hist * emb_dim;           // 3200 floats per output row
    const int batch   = out_size / rowlen;        // 16384

    const size_t shmem = (size_t)rowlen * sizeof(float);  // 12.8 KB / block (async path staging)

    remap_rows_kernel<<<batch, BLOCK, shmem, stream>>>(
        emb, pos, n_valid, emb_dim, rowlen, (float*)d_out);
}